// GatedGCN_layer_3058016715104
// MI455X (gfx1250) — compile-verified
//
#include <hip/hip_runtime.h>
#include <hip/hip_bf16.h>

#define Nn 100000
#define Mm 500000
#define Dd 128
#define BN_EPS 1e-5f

typedef __attribute__((ext_vector_type(2))) float v2f;
typedef __attribute__((ext_vector_type(8))) float v8f;

// ---------------------------------------------------------------------------
// 16x128 GEMM tile: out[16,128] = A[16,128] @ W[128,128]^T + bias, fp32 WMMA.
// A-frag (16x4 f32): lane l holds M=l&15, K = 2*(l>>4) + {0,1}
// B-frag (4x16 f32): lane l holds N=l&15, K = 2*(l>>4) + {0,1}; B[k][n]=W[n][k]
// C/D   (16x16 f32): lane l, vgpr r holds M = r + 8*(l>>4), N = l&15
// ---------------------------------------------------------------------------
__device__ __forceinline__ void gemm_tile_16x128(
    const float* __restrict__ Arow0,  // 16 rows x 128 cols, row-major
    const float* __restrict__ W,      // [128 out, 128 in] row-major
    const float* __restrict__ bias,   // [128]
    int lane, v8f acc[8]) {
  const int lh = lane & 15;
  const int kh = (lane >> 4) << 1;  // 0 or 2
  const float* ap = Arow0 + lh * Dd + kh;
#pragma unroll 2
  for (int k0 = 0; k0 < Dd; k0 += 4) {
    v2f a = *(const v2f*)(ap + k0);
#pragma unroll
    for (int nt = 0; nt < 8; ++nt) {
      v2f b = *(const v2f*)(W + (nt * 16 + lh) * Dd + k0 + kh);
      acc[nt] = __builtin_amdgcn_wmma_f32_16x16x4_f32(
          false, a, false, b, (short)0, acc[nt], false, false);
    }
  }
#pragma unroll
  for (int nt = 0; nt < 8; ++nt) {
    float bv = bias[nt * 16 + lh];
#pragma unroll
    for (int r = 0; r < 8; ++r) acc[nt][r] += bv;
  }
}

__device__ __forceinline__ void zero_acc(v8f acc[8]) {
  v8f z;
#pragma unroll
  for (int r = 0; r < 8; ++r) z[r] = 0.f;
#pragma unroll
  for (int i = 0; i < 8; ++i) acc[i] = z;
}

// ---------------------------------------------------------------------------
// K1: node projections. 4 waves/block; wave w computes one of AX/BX/DX/EX for
// the block's 16-row tile. N = 6250 * 16 exactly -> full EXEC everywhere.
// ---------------------------------------------------------------------------
__global__ __launch_bounds__(128) void node_gemm_k(
    const float* __restrict__ X,
    const float* __restrict__ WA, const float* __restrict__ bA,
    const float* __restrict__ WB, const float* __restrict__ bB,
    const float* __restrict__ WD, const float* __restrict__ bD,
    const float* __restrict__ WE, const float* __restrict__ bE,
    float* __restrict__ AX, float* __restrict__ BX,
    float* __restrict__ DX, float* __restrict__ EXo) {
  const int wave = threadIdx.x >> 5;
  const int lane = threadIdx.x & 31;
  const int m0 = blockIdx.x * 16;
  const float *W, *b;
  float* out;
  if (wave == 0)      { W = WA; b = bA; out = AX; }
  else if (wave == 1) { W = WB; b = bB; out = BX; }
  else if (wave == 2) { W = WD; b = bD; out = DX; }
  else                { W = WE; b = bE; out = EXo; }
  v8f acc[8];
  zero_acc(acc);
  gemm_tile_16x128(X + (size_t)m0 * Dd, W, b, lane, acc);
  const int lh = lane & 15, hh = lane >> 4;
#pragma unroll
  for (int nt = 0; nt < 8; ++nt)
#pragma unroll
    for (int r = 0; r < 8; ++r)
      out[(size_t)(m0 + r + 8 * hh) * Dd + nt * 16 + lh] = acc[nt][r];
}

// ---------------------------------------------------------------------------
// K2: edge projection CE + message. 2 waves/block, each a 16-edge tile.
// e = CE + DX[src] + EX[dst]; sig = sigmoid(e); scatter-add num/den by dst.
// ---------------------------------------------------------------------------
__global__ __launch_bounds__(64) void edge_msg_k(
    const float* __restrict__ EXin,
    const float* __restrict__ WC, const float* __restrict__ bC,
    const float* __restrict__ BX, const float* __restrict__ DX,
    const float* __restrict__ EXn,
    const int* __restrict__ src, const int* __restrict__ dst,
    float* __restrict__ e_out, float* __restrict__ num,
    float* __restrict__ den) {
  const int wave = threadIdx.x >> 5;
  const int lane = threadIdx.x & 31;
  const int m0 = (blockIdx.x * 2 + wave) * 16;
  v8f acc[8];
  zero_acc(acc);
  gemm_tile_16x128(EXin + (size_t)m0 * Dd, WC, bC, lane, acc);
  const int lh = lane & 15, hh = lane >> 4;
  int sA[8], dA[8];
#pragma unroll
  for (int r = 0; r < 8; ++r) {
    const int em = m0 + r + 8 * hh;  // uniform within half-wave -> broadcast
    sA[r] = src[em];
    dA[r] = dst[em];
  }
#pragma unroll
  for (int nt = 0; nt < 8; ++nt) {
    const int n = nt * 16 + lh;
#pragma unroll
    for (int r = 0; r < 8; ++r) {
      const int em = m0 + r + 8 * hh;
      float ev = acc[nt][r] + DX[(size_t)sA[r] * Dd + n] +
                 EXn[(size_t)dA[r] * Dd + n];
      e_out[(size_t)em * Dd + n] = ev;
      float sg = 1.f / (1.f + __expf(-ev));
      float bx = BX[(size_t)sA[r] * Dd + n];
      __hip_atomic_fetch_add(&num[(size_t)dA[r] * Dd + n], sg * bx,
                             __ATOMIC_RELAXED, __HIP_MEMORY_SCOPE_AGENT);
      __hip_atomic_fetch_add(&den[(size_t)dA[r] * Dd + n], sg,
                             __ATOMIC_RELAXED, __HIP_MEMORY_SCOPE_AGENT);
    }
  }
}

// ---------------------------------------------------------------------------
__global__ void zero_k(float* __restrict__ p, long n) {
  long i = (long)blockIdx.x * blockDim.x + threadIdx.x;
  if (i < n) p[i] = 0.f;
}

// K3: H_pre = where(den>0, AX + num/den, X) * snorm_n ; column sums/sumsq.
__global__ __launch_bounds__(128) void node_stats_k(
    const float* __restrict__ X, const float* __restrict__ AX,
    const float* __restrict__ num, const float* __restrict__ den,
    const float* __restrict__ snorm_n, float* __restrict__ Hpre,
    float* __restrict__ statH) {
  const int c = threadIdx.x;
  const int r0 = blockIdx.x * 100;
  float s = 0.f, ss = 0.f;
  for (int i = 0; i < 100; ++i) {
    const int row = r0 + i;
    const size_t idx = (size_t)row * Dd + c;
    float dv = den[idx];
    float h = dv > 0.f ? AX[idx] + num[idx] / dv : X[idx];
    h *= snorm_n[row];
    Hpre[idx] = h;
    s += h;
    ss += h * h;
  }
  __hip_atomic_fetch_add(&statH[c], s, __ATOMIC_RELAXED, __HIP_MEMORY_SCOPE_AGENT);
  __hip_atomic_fetch_add(&statH[Dd + c], ss, __ATOMIC_RELAXED, __HIP_MEMORY_SCOPE_AGENT);
}

// K4: column sums/sumsq of e * snorm_e.
__global__ __launch_bounds__(128) void edge_stats_k(
    const float* __restrict__ e, const float* __restrict__ snorm_e,
    float* __restrict__ statE) {
  const int c = threadIdx.x;
  const int r0 = blockIdx.x * 125;
  float s = 0.f, ss = 0.f;
  for (int i = 0; i < 125; ++i) {
    const int row = r0 + i;
    const size_t idx = (size_t)row * Dd + c;
    float v = e[idx] * snorm_e[row];
    s += v;
    ss += v * v;
  }
  __hip_atomic_fetch_add(&statE[c], s, __ATOMIC_RELAXED, __HIP_MEMORY_SCOPE_AGENT);
  __hip_atomic_fetch_add(&statE[Dd + c], ss, __ATOMIC_RELAXED, __HIP_MEMORY_SCOPE_AGENT);
}

// K5: fold BN stats into per-column scale/shift for H and E.
__global__ __launch_bounds__(256) void finalize_k(
    const float* __restrict__ statH, const float* __restrict__ statE,
    const float* __restrict__ gamma_h, const float* __restrict__ beta_h,
    const float* __restrict__ gamma_e, const float* __restrict__ beta_e,
    float* __restrict__ coefH, float* __restrict__ coefE) {
  const int t = threadIdx.x;
  if (t < Dd) {
    float mean = statH[t] / (float)Nn;
    float var = statH[Dd + t] / (float)Nn - mean * mean;
    float sc = rsqrtf(var + BN_EPS) * gamma_h[t];
    coefH[t] = sc;
    coefH[Dd + t] = beta_h[t] - mean * sc;
  } else {
    const int c = t - Dd;
    float mean = statE[c] / (float)Mm;
    float var = statE[Dd + c] / (float)Mm - mean * mean;
    float sc = rsqrtf(var + BN_EPS) * gamma_e[c];
    coefE[c] = sc;
    coefE[Dd + c] = beta_e[c] - mean * sc;
  }
}

// K6: H_out = X + relu(Hpre*scale + shift)
__global__ __launch_bounds__(256) void apply_h_k(
    const float* __restrict__ X, const float* __restrict__ Hpre,
    const float* __restrict__ coefH, float* __restrict__ outH) {
  long i = (long)blockIdx.x * 256 + threadIdx.x;
  if (i >= (long)Nn * Dd) return;
  const int c = (int)(i & (Dd - 1));
  float v = Hpre[i] * coefH[c] + coefH[Dd + c];
  outH[i] = X[i] + fmaxf(v, 0.f);
}

// K7: E_out = E_X + relu((e*snorm_e)*scale + shift)
__global__ __launch_bounds__(256) void apply_e_k(
    const float* __restrict__ EXin, const float* __restrict__ e,
    const float* __restrict__ snorm_e, const float* __restrict__ coefE,
    float* __restrict__ outE) {
  long i = (long)blockIdx.x * 256 + threadIdx.x;
  if (i >= (long)Mm * Dd) return;
  const int c = (int)(i & (Dd - 1));
  const long row = i >> 7;
  float v = e[i] * snorm_e[row] * coefE[c] + coefE[Dd + c];
  outE[i] = EXin[i] + fmaxf(v, 0.f);
}

// ---------------------------------------------------------------------------
extern "C" void kernel_launch(void* const* d_in, const int* in_sizes, int n_in,
                              void* d_out, int out_size, void* d_ws,
                              size_t ws_size, hipStream_t stream) {
  const float* X       = (const float*)d_in[0];
  const float* E_X     = (const float*)d_in[1];
  const float* snorm_n = (const float*)d_in[2];
  const float* snorm_e = (const float*)d_in[3];
  const int*   src     = (const int*)d_in[4];
  const int*   dst     = (const int*)d_in[5];
  const float* WA = (const float*)d_in[6],  *bA = (const float*)d_in[7];
  const float* WB = (const float*)d_in[8],  *bB = (const float*)d_in[9];
  const float* WC = (const float*)d_in[10], *bC = (const float*)d_in[11];
  const float* WD = (const float*)d_in[12], *bD = (const float*)d_in[13];
  const float* WE = (const float*)d_in[14], *bE = (const float*)d_in[15];
  const float* gamma_h = (const float*)d_in[16], *beta_h = (const float*)d_in[17];
  const float* gamma_e = (const float*)d_in[18], *beta_e = (const float*)d_in[19];

  const size_t ND = (size_t)Nn * Dd;  // 12.8M
  const size_t MD = (size_t)Mm * Dd;  // 64M
  float* ws    = (float*)d_ws;
  float* AX    = ws;
  float* BX    = AX + ND;
  float* DX    = BX + ND;
  float* EXb   = DX + ND;
  float* ebuf  = EXb + ND;
  float* num   = ebuf + MD;       // ---- zeroed region start
  float* den   = num + ND;
  float* statH = den + ND;        // 256 floats (sum | sumsq)
  float* statE = statH + 2 * Dd;  // 256 floats ---- zeroed region end
  float* Hpre  = statE + 2 * Dd + 512;
  float* coefH = Hpre + ND;       // 256 floats (scale | shift)
  float* coefE = coefH + 2 * Dd;  // 256 floats

  float* outH = (float*)d_out;
  float* outE = outH + ND;

  const long zn = (long)(2 * ND + 4 * Dd);
  zero_k<<<(unsigned)((zn + 255) / 256), 256, 0, stream>>>(num, zn);

  node_gemm_k<<<Nn / 16, 128, 0, stream>>>(X, WA, bA, WB, bB, WD, bD, WE, bE,
                                           AX, BX, DX, EXb);
  edge_msg_k<<<Mm / 32, 64, 0, stream>>>(E_X, WC, bC, BX, DX, EXb, src, dst,
                                         ebuf, num, den);
  node_stats_k<<<Nn / 100, 128, 0, stream>>>(X, AX, num, den, snorm_n, Hpre,
                                             statH);
  edge_stats_k<<<Mm / 125, 128, 0, stream>>>(ebuf, snorm_e, statE);
  finalize_k<<<1, 256, 0, stream>>>(statH, statE, gamma_h, beta_h, gamma_e,
                                    beta_e, coefH, coefE);
  apply_h_k<<<(unsigned)((ND + 255) / 256), 256, 0, stream>>>(X, Hpre, coefH,
                                                              outH);
  apply_e_k<<<(unsigned)((MD + 255) / 256), 256, 0, stream>>>(E_X, ebuf,
                                                              snorm_e, coefE,
                                                              outE);
}